// RecursiveMobiusLayer_56057913147853
// MI455X (gfx1250) — compile-verified
//
#include <hip/hip_runtime.h>
#include <stdint.h>

// 1-D periodic complex stencil, `cycles` (<=32) fused time steps via LDS
// temporal tiling with a fixed compute trapezoid and register-resident centers.
// f_{t+1} = f + DT*( f[i-1]+f[i+1] - (NONLIN*|f|^2 + DAMP)*f )

#define TILE     2048        // complex elements per block (N = 2^23 divides evenly)
#define HMAX     32          // max supported halo (= max fused cycles); ref uses 20
#define NTHREADS 256
#define K        9           // elements per thread; region = K*NTHREADS = 2304
#define LO       1           // compute region = [LO, LO + K*NTHREADS) = [1, 2305)
#define LPITCH   2308        // float2 per LDS buffer (>= 2306; keeps buf1 16B-aligned)

#define DEPTH  0
#define NONLIN (0.1f * (1.0f + DEPTH * 0.2f))   // 0.1
#define DAMP   (0.05f * (1.0f - DEPTH * 0.1f))  // 0.05
#define DT     0.01f

__global__ __launch_bounds__(NTHREADS)
void nls_fused_kernel(const float2* __restrict__ f_in,
                      const int*    __restrict__ cyc,
                      float2*       __restrict__ f_out,
                      int n)
{
    extern __shared__ float2 sm[];       // [2][LPITCH]; dynamic => LDS byte offset 0
    float2* buf0 = sm;
    float2* buf1 = sm + LPITCH;

    const int tid = threadIdx.x;
    const int g0  = (int)blockIdx.x * TILE;      // global index of interior start

    int h = cyc[0];                               // fused steps == halo width
    h = h < 0 ? 0 : (h > HMAX ? HMAX : h);

    // ---- Stage tile + halo into LDS via async global->LDS copies.
    // Per-lane addressing makes the periodic wraparound free.
    // Buffer index convention: LDS index i <-> global index g0 + (i - HMAX).
    const int load_n = TILE + 2 * h;
    for (int j = tid; j < load_n; j += NTHREADS) {
        int gidx = g0 - h + j;
        if (gidx < 0)  gidx += n;
        if (gidx >= n) gidx -= n;
        uint32_t lds_off = (uint32_t)((HMAX - h + j) * (int)sizeof(float2));
        unsigned long long gaddr =
            (unsigned long long)(uintptr_t)(f_in + gidx);
        asm volatile("global_load_async_to_lds_b64 %0, %1, off"
                     :: "v"(lds_off), "v"(gaddr) : "memory");
    }
    asm volatile("s_wait_asynccnt 0x0" ::: "memory");
    __syncthreads();

    // ---- Center values live in registers; LDS only serves neighbor exchange.
    float2 c[K];
    #pragma unroll
    for (int k = 0; k < K; ++k)
        c[k] = buf0[LO + tid + k * NTHREADS];

    // ---- Fixed-region fused steps (ping-pong LDS). Stale boundary data
    // contaminates inward at 1 elem/step; after h steps the interior
    // [HMAX, HMAX+TILE) is exactly the surviving correct region.
    float2* cur = buf0;
    float2* nxt = buf1;
    for (int s = 0; s < h; ++s) {
        #pragma unroll
        for (int k = 0; k < K; ++k) {
            const int i = LO + tid + k * NTHREADS;
            float2 l  = cur[i - 1];
            float2 r  = cur[i + 1];
            float2 cc = c[k];
            float mag2 = fmaf(cc.x, cc.x, cc.y * cc.y);
            float coef = fmaf(NONLIN, mag2, DAMP);
            float tx = fmaf(-coef, cc.x, l.x + r.x);
            float ty = fmaf(-coef, cc.y, l.y + r.y);
            float2 o;
            o.x = fmaf(DT, tx, cc.x);
            o.y = fmaf(DT, ty, cc.y);
            nxt[i] = o;                  // publish for neighbors' next step
            c[k]   = o;                  // own center stays in registers
        }
        __syncthreads();                 // publish nxt; guards WAR on cur
        float2* t = cur; cur = nxt; nxt = t;
    }

    // ---- Interior results are already in registers: direct coalesced stores.
    #pragma unroll
    for (int k = 0; k < K; ++k) {
        const int i = LO + tid + k * NTHREADS;
        if (i >= HMAX && i < HMAX + TILE)      // k=1..7 always true; k=0,8 edge
            f_out[g0 + (i - HMAX)] = c[k];
    }
}

extern "C" void kernel_launch(void* const* d_in, const int* in_sizes, int n_in,
                              void* d_out, int out_size, void* d_ws, size_t ws_size,
                              hipStream_t stream)
{
    const float2* f_in  = (const float2*)d_in[0];  // [N,2] f32 == N float2
    const int*    cyc   = (const int*)d_in[1];     // scalar cycles (device)
    float2*       f_out = (float2*)d_out;

    const int n = in_sizes[0] / 2;                 // complex element count
    const int nblocks = n / TILE;                  // N = 2^23 -> 4096 blocks
    const size_t smem = (size_t)2 * LPITCH * sizeof(float2);   // 36,928 B

    nls_fused_kernel<<<nblocks, NTHREADS, smem, stream>>>(f_in, cyc, f_out, n);
}